// NgcfConv_49581102465505
// MI455X (gfx1250) — compile-verified
//
#include <hip/hip_runtime.h>

#define N_NODES 100000
#define D 64
#define ROWS_PER_BLK 32
#define APAD 65   // LDS row stride (floats) to avoid 64-bank conflicts

typedef __attribute__((ext_vector_type(2))) float v2f;
typedef __attribute__((ext_vector_type(8))) float v8f;

// ---------------- kernel 1: zero agg workspace ----------------
__global__ __launch_bounds__(256) void zero_kernel(float4* __restrict__ p, int n4) {
    int i = blockIdx.x * blockDim.x + threadIdx.x;
    int stride = gridDim.x * blockDim.x;
    for (; i < n4; i += stride) p[i] = make_float4(0.f, 0.f, 0.f, 0.f);
}

// ---------------- kernel 2: SpMM scatter-add (one wave32 per edge) ----------------
__global__ __launch_bounds__(256) void spmm_kernel(const int* __restrict__ erow,
                                                   const int* __restrict__ ecol,
                                                   const float* __restrict__ evals,
                                                   const float* __restrict__ feat,
                                                   float* __restrict__ agg,
                                                   int nEdges) {
    int wave = (int)((blockIdx.x * (unsigned)blockDim.x + threadIdx.x) >> 5);
    int lane = threadIdx.x & 31;
    if (wave >= nEdges) return;
    int r = erow[wave];
    int c = ecol[wave];
    float v = evals[wave];
    const float* src = feat + (size_t)c * D;
    float*       dst = agg  + (size_t)r * D;
    // lanes cover features [0..31] and [32..63]: fully coalesced f32 atomics
    atomicAdd(&dst[lane],      v * src[lane]);
    atomicAdd(&dst[lane + 32], v * src[lane + 32]);
}

// ---------------- kernel 3: dense part with FP32 WMMA ----------------
// out = leaky(agg @ W1 + b1) + leaky((agg*feat) @ W2 + b2)
// Block: 256 threads = 8 waves; covers 32 rows x 64 cols of output.
// Wave (wy,wx): wy in {0,1} picks 16-row band, wx in {0..3} picks 16-col band.
__global__ __launch_bounds__(256) void dense_kernel(const float* __restrict__ agg,
                                                    const float* __restrict__ feat,
                                                    const float* __restrict__ W1,
                                                    const float* __restrict__ b1,
                                                    const float* __restrict__ W2,
                                                    const float* __restrict__ b2,
                                                    float* __restrict__ out) {
    __shared__ float sA1[ROWS_PER_BLK * APAD];   // agg tile
    __shared__ float sA2[ROWS_PER_BLK * APAD];   // (agg * feat) tile
    __shared__ float sW1[D * APAD];
    __shared__ float sW2[D * APAD];

    const int tid = threadIdx.x;
    const int rowBase = blockIdx.x * ROWS_PER_BLK;   // N_NODES % 32 == 0: no tail

    // Stage A tiles (coalesced: consecutive threads -> consecutive features)
    #pragma unroll
    for (int i = 0; i < (ROWS_PER_BLK * D) / 256; ++i) {
        int e = tid + i * 256;
        int r = e >> 6, c = e & 63;
        float a = agg [(size_t)(rowBase + r) * D + c];
        float f = feat[(size_t)(rowBase + r) * D + c];
        sA1[r * APAD + c] = a;
        sA2[r * APAD + c] = a * f;
    }
    // Stage weights (64x64 each)
    #pragma unroll
    for (int i = 0; i < (D * D) / 256; ++i) {
        int e = tid + i * 256;
        int r = e >> 6, c = e & 63;
        sW1[r * APAD + c] = W1[e];
        sW2[r * APAD + c] = W2[e];
    }
    __syncthreads();

    const int wave = tid >> 5;
    const int lane = tid & 31;
    const int wy = wave >> 2;          // 0..1 : row band
    const int wx = wave & 3;           // 0..3 : col band
    const int l15   = lane & 15;       // A: row-in-tile (M); B: col-in-tile (N)
    const int khalf = (lane >> 4) * 2; // K sub-offset per ISA fp32 A/B layout

    const int arow = wy * 16 + l15;    // A-matrix row within the 32-row tile
    const int bcol = wx * 16 + l15;    // B-matrix column within 0..63

    v8f c1 = {};  // accumulator for agg @ W1
    v8f c2 = {};  // accumulator for (agg*feat) @ W2

    #pragma unroll
    for (int kk = 0; kk < D; kk += 4) {
        const int k = kk + khalf;
        // A operand (16x4 fp32): lane holds row M=l15; VGPR0/1 = K = k, k+1
        v2f a1, a2, w1v, w2v;
        a1.x = sA1[arow * APAD + k];
        a1.y = sA1[arow * APAD + k + 1];
        a2.x = sA2[arow * APAD + k];
        a2.y = sA2[arow * APAD + k + 1];
        // B operand (4x16 fp32): lane holds col N=l15; VGPR0/1 = K = k, k+1
        w1v.x = sW1[(k    ) * APAD + bcol];
        w1v.y = sW1[(k + 1) * APAD + bcol];
        w2v.x = sW2[(k    ) * APAD + bcol];
        w2v.y = sW2[(k + 1) * APAD + bcol];
        // 8 args: (neg_a, A, neg_b, B, c_mod, C, reuse_a, reuse_b)
        c1 = __builtin_amdgcn_wmma_f32_16x16x4_f32(false, a1, false, w1v,
                                                   (short)0, c1, false, false);
        c2 = __builtin_amdgcn_wmma_f32_16x16x4_f32(false, a2, false, w2v,
                                                   (short)0, c2, false, false);
    }

    // Epilogue: bias + leaky ReLU + sum. C/D layout: VGPR j -> row j (lanes 0-15)
    // or row j+8 (lanes 16-31); column = l15 within the wx band.
    const int col = wx * 16 + l15;
    const float bias1 = b1[col];
    const float bias2 = b2[col];
    const int rofs = (lane < 16) ? 0 : 8;
    #pragma unroll
    for (int j = 0; j < 8; ++j) {
        int r = rowBase + wy * 16 + j + rofs;
        float x1 = c1[j] + bias1;
        x1 = (x1 >= 0.f) ? x1 : 0.2f * x1;
        float x2 = c2[j] + bias2;
        x2 = (x2 >= 0.f) ? x2 : 0.2f * x2;
        out[(size_t)r * D + col] = x1 + x2;
    }
}

extern "C" void kernel_launch(void* const* d_in, const int* in_sizes, int n_in,
                              void* d_out, int out_size, void* d_ws, size_t ws_size,
                              hipStream_t stream) {
    const int*   edge_row  = (const int*)  d_in[0];
    const int*   edge_col  = (const int*)  d_in[1];
    const float* edge_vals = (const float*)d_in[2];
    const float* node_feat = (const float*)d_in[3];
    const float* W1        = (const float*)d_in[4];
    const float* b1        = (const float*)d_in[5];
    const float* W2        = (const float*)d_in[6];
    const float* b2        = (const float*)d_in[7];
    float* out = (float*)d_out;

    const int nEdges = in_sizes[0];
    const int nNodes = in_sizes[3] / D;

    float* agg = (float*)d_ws;   // nNodes * D floats of scratch

    // 1) zero the aggregation buffer (must redo every call)
    {
        int n4 = (nNodes * D) / 4;
        int blocks = (n4 + 255) / 256;
        if (blocks > 4096) blocks = 4096;  // grid-stride
        zero_kernel<<<blocks, 256, 0, stream>>>((float4*)agg, n4);
    }
    // 2) SpMM scatter-add: one wave32 per edge
    {
        int wavesPerBlk = 256 / 32;
        int blocks = (nEdges + wavesPerBlk - 1) / wavesPerBlk;
        spmm_kernel<<<blocks, 256, 0, stream>>>(edge_row, edge_col, edge_vals,
                                                node_feat, agg, nEdges);
    }
    // 3) dense WMMA epilogue: 32 rows per block
    {
        int blocks = nNodes / ROWS_PER_BLK;   // 100000 / 32 = 3125 exact
        dense_kernel<<<blocks, 256, 0, stream>>>(agg, node_feat, W1, b1, W2, b2, out);
    }
}